// GCN_12317966204981
// MI455X (gfx1250) — compile-verified
//
#include <hip/hip_runtime.h>
#include <math.h>

// Problem constants (match reference setup_inputs)
#define NNODES 100000
#define NEDGES 1600000
#define NGRAPH 512
#define FIN    128
#define HID    64
#define NCLS   10
#define BN_EPS 1e-5f

typedef float v2f __attribute__((ext_vector_type(2)));
typedef float v8f __attribute__((ext_vector_type(8)));

// Native f32 atomic add (global_atomic_add_f32) — avoid CAS-loop fallback.
__device__ __forceinline__ void atomAdd(float* p, float v) { unsafeAtomicAdd(p, v); }

// ---------------------------------------------------------------------------
// GEMM: Y[N,64] = X[N,K] @ W[K,64] using V_WMMA_F32_16X16X4_F32 (full fp32).
// One wave per 16x16 output tile. nrows % 16 == 0, K % 4 == 0 guaranteed.
// ---------------------------------------------------------------------------
__global__ void gemm_wmma_f32(const float* __restrict__ X,
                              const float* __restrict__ W,
                              float* __restrict__ Y,
                              int nrows, int K) {
  const int lane = threadIdx.x & 31;
  const int wave = threadIdx.x >> 5;
  const int gw   = blockIdx.x * 8 + wave;       // 8 waves / 256-thread block
  const int tileM = gw >> 2;                    // HID/16 == 4 tiles across
  const int tileN = gw & 3;
  if (tileM * 16 >= nrows) return;              // wave-uniform; EXEC stays full

  const int half = lane >> 4;                   // 0: K=k..k+1, 1: K=k+2..k+3
  const int l15  = lane & 15;
  const int row  = tileM * 16 + l15;            // A: M striped over lanes
  const int col  = tileN * 16 + l15;            // B: N striped over lanes

  v8f acc = {};
  const float* xr = X + (size_t)row * K;
  for (int k = 0; k < K; k += 4) {
    const int kk = k + 2 * half;
    v2f a = *(const v2f*)(xr + kk);                       // A[row][kk..kk+1]
    v2f b;
    b.x = W[(size_t)kk * HID + col];                      // B[kk][col]
    b.y = W[(size_t)(kk + 1) * HID + col];                // B[kk+1][col]
    acc = __builtin_amdgcn_wmma_f32_16x16x4_f32(
        /*neg_a=*/false, a, /*neg_b=*/false, b,
        /*c_mod=*/(short)0, acc, /*reuse_a=*/false, /*reuse_b=*/false);
  }
  // C/D layout: VGPR r -> rows (r, r+8) split across half-waves
  #pragma unroll
  for (int r = 0; r < 8; ++r) {
    const int orow = tileM * 16 + r + 8 * half;
    Y[(size_t)orow * HID + col] = acc[r];
  }
}

// ---------------------------------------------------------------------------
// Graph preprocessing (done once; shared by all 4 layers)
// ---------------------------------------------------------------------------
__global__ void zero_k(float* __restrict__ p, int n) {
  int t = blockIdx.x * blockDim.x + threadIdx.x;
  if (t < n) p[t] = 0.0f;
}

__global__ void deg_k(const int* __restrict__ dst, float* __restrict__ deg) {
  int e = blockIdx.x * blockDim.x + threadIdx.x;
  if (e < NEDGES) atomAdd(&deg[dst[e]], 1.0f);
}

__global__ void dis_k(float* __restrict__ deg) {  // in-place: deg -> deg^-1/2
  int n = blockIdx.x * blockDim.x + threadIdx.x;
  if (n < NNODES) deg[n] = rsqrtf(deg[n] + 1.0f);  // +1 for self-loop
}

__global__ void coef_k(const int* __restrict__ src, const int* __restrict__ dst,
                       const float* __restrict__ dis, float* __restrict__ coef) {
  int e = blockIdx.x * blockDim.x + threadIdx.x;
  if (e < NEDGES) coef[e] = dis[src[e]] * dis[dst[e]];
}

// ---------------------------------------------------------------------------
// GCN aggregation: agg = self-loop term + bias (init), then edge scatter-add.
// ---------------------------------------------------------------------------
__global__ void agg_init_k(const float* __restrict__ h, const float* __restrict__ dis,
                           const float* __restrict__ bias, float* __restrict__ agg) {
  int t = blockIdx.x * blockDim.x + threadIdx.x;
  if (t >= NNODES * HID) return;
  int n = t >> 6, f = t & 63;
  float d = dis[n];
  agg[t] = h[t] * (d * d) + bias[f];
}

// 16 lanes per edge, 4 features each: coalesced 256B gather of h[src], then
// 4 native f32 atomics into agg[dst]. All L2-resident on MI455X (192MB L2).
__global__ void agg_edges_k(const float* __restrict__ h, const int* __restrict__ src,
                            const int* __restrict__ dst, const float* __restrict__ coef,
                            float* __restrict__ agg) {
  int t = blockIdx.x * blockDim.x + threadIdx.x;
  if (t >= NEDGES * 16) return;
  int e  = t >> 4;
  int fq = (t & 15) << 2;
  float c = coef[e];
  const float4 hv = *(const float4*)(h + (size_t)src[e] * HID + fq);
  float* d = agg + (size_t)dst[e] * HID + fq;
  atomAdd(d + 0, hv.x * c);
  atomAdd(d + 1, hv.y * c);
  atomAdd(d + 2, hv.z * c);
  atomAdd(d + 3, hv.w * c);
}

// ---------------------------------------------------------------------------
// BatchNorm (training-mode batch stats) + ReLU
// ---------------------------------------------------------------------------
__global__ void bn_stats_k(const float* __restrict__ h, float* __restrict__ stats,
                           int n) {  // stats[0:64]=sum, stats[64:128]=sumsq (pre-zeroed)
  __shared__ float s_sum[256];
  __shared__ float s_sq[256];
  const int f = threadIdx.x & 63;
  const int r = threadIdx.x >> 6;  // 0..3
  float sum = 0.0f, sq = 0.0f;
  for (int row = blockIdx.x * 4 + r; row < n; row += gridDim.x * 4) {
    float v = h[(size_t)row * HID + f];
    sum += v; sq += v * v;
  }
  s_sum[threadIdx.x] = sum;
  s_sq[threadIdx.x]  = sq;
  __syncthreads();
  if (threadIdx.x < 64) {
    float ts = s_sum[f] + s_sum[64 + f] + s_sum[128 + f] + s_sum[192 + f];
    float tq = s_sq[f]  + s_sq[64 + f]  + s_sq[128 + f]  + s_sq[192 + f];
    atomAdd(&stats[f], ts);
    atomAdd(&stats[64 + f], tq);
  }
}

__global__ void bn_coeffs_k(const float* __restrict__ stats,
                            const float* __restrict__ gamma,
                            const float* __restrict__ beta,
                            float* __restrict__ coeffs) {
  int f = threadIdx.x;
  if (f < HID) {
    const float invN = 1.0f / (float)NNODES;
    float mu  = stats[f] * invN;
    float var = stats[64 + f] * invN - mu * mu;   // biased variance
    float sc  = gamma[f] * rsqrtf(var + BN_EPS);
    coeffs[f]       = sc;
    coeffs[64 + f]  = beta[f] - mu * sc;
  }
}

__global__ void bn_apply_k(float* __restrict__ h, const float* __restrict__ coeffs) {
  int t = blockIdx.x * blockDim.x + threadIdx.x;
  if (t >= NNODES * HID) return;
  int f = t & 63;
  h[t] = fmaxf(h[t] * coeffs[f] + coeffs[64 + f], 0.0f);
}

// ---------------------------------------------------------------------------
// Global mean pool + FC + log_softmax
// ---------------------------------------------------------------------------
__global__ void pool_k(const float* __restrict__ h, const int* __restrict__ batch,
                       float* __restrict__ psum, float* __restrict__ pcnt) {
  int t = blockIdx.x * blockDim.x + threadIdx.x;
  if (t >= NNODES * 16) return;
  int nidx = t >> 4;
  int fq   = (t & 15) << 2;
  int g    = batch[nidx];
  const float4 hv = *(const float4*)(h + (size_t)nidx * HID + fq);
  float* d = psum + (size_t)g * HID + fq;
  atomAdd(d + 0, hv.x);
  atomAdd(d + 1, hv.y);
  atomAdd(d + 2, hv.z);
  atomAdd(d + 3, hv.w);
  if ((t & 15) == 0) atomAdd(&pcnt[g], 1.0f);
}

__global__ void fc_logsoftmax_k(const float* __restrict__ psum,
                                const float* __restrict__ pcnt,
                                const float* __restrict__ fcW,
                                const float* __restrict__ fcb,
                                float* __restrict__ out) {
  int g = blockIdx.x * blockDim.x + threadIdx.x;
  if (g >= NGRAPH) return;
  float inv = 1.0f / fmaxf(pcnt[g], 1.0f);
  float logits[NCLS];
  #pragma unroll
  for (int c = 0; c < NCLS; ++c) logits[c] = fcb[c];
  for (int f = 0; f < HID; ++f) {
    float v = psum[(size_t)g * HID + f] * inv;
    #pragma unroll
    for (int c = 0; c < NCLS; ++c) logits[c] += v * fcW[f * NCLS + c];
  }
  float m = logits[0];
  #pragma unroll
  for (int c = 1; c < NCLS; ++c) m = fmaxf(m, logits[c]);
  float se = 0.0f;
  #pragma unroll
  for (int c = 0; c < NCLS; ++c) se += expf(logits[c] - m);
  float lse = m + logf(se);
  #pragma unroll
  for (int c = 0; c < NCLS; ++c) out[g * NCLS + c] = logits[c] - lse;
}

// ---------------------------------------------------------------------------
// Host launch
// ---------------------------------------------------------------------------
extern "C" void kernel_launch(void* const* d_in, const int* in_sizes, int n_in,
                              void* d_out, int out_size, void* d_ws, size_t ws_size,
                              hipStream_t stream) {
  (void)in_sizes; (void)n_in; (void)out_size; (void)ws_size;
  const float* x     = (const float*)d_in[0];
  const int*   ei    = (const int*)d_in[1];        // [2, E]: row0=src, row1=dst
  const int*   batch = (const int*)d_in[2];
  const float* W[4]  = {(const float*)d_in[3], (const float*)d_in[7],
                        (const float*)d_in[11], (const float*)d_in[15]};
  const float* b[4]  = {(const float*)d_in[4], (const float*)d_in[8],
                        (const float*)d_in[12], (const float*)d_in[16]};
  const float* g[4]  = {(const float*)d_in[5], (const float*)d_in[9],
                        (const float*)d_in[13], (const float*)d_in[17]};
  const float* bt[4] = {(const float*)d_in[6], (const float*)d_in[10],
                        (const float*)d_in[14], (const float*)d_in[18]};
  const float* fcW = (const float*)d_in[19];
  const float* fcb = (const float*)d_in[20];
  float* out = (float*)d_out;

  const int* e_src = ei;
  const int* e_dst = ei + NEDGES;

  // Workspace layout (~58 MB of f32)
  float* ws     = (float*)d_ws;
  float* dis    = ws;                                 // [N]  deg -> deg^-1/2
  float* coef   = dis + NNODES;                       // [E]
  float* bufA   = coef + NEDGES;                      // [N*64] gemm output
  float* bufB   = bufA + (size_t)NNODES * HID;        // [N*64] activations
  float* stats  = bufB + (size_t)NNODES * HID;        // [128]
  float* coeffs = stats + 128;                        // [128]
  float* psum   = coeffs + 128;                       // [G*64]
  float* pcnt   = psum + (size_t)NGRAPH * HID;        // [G]

  const int elemN   = NNODES * HID;                   // 6.4M
  const int gElem   = (elemN + 255) / 256;
  const int gEdge   = (NEDGES + 255) / 256;
  const int gEdge16 = (NEDGES * 16 + 255) / 256;
  const int gNode   = (NNODES + 255) / 256;
  const int gNode16 = (NNODES * 16 + 255) / 256;
  const int nTiles  = (NNODES / 16) * (HID / 16);     // 25000
  const int gGemm   = (nTiles + 7) / 8;               // 8 waves per block

  // --- graph normalization (once) ---
  zero_k<<<gNode, 256, 0, stream>>>(dis, NNODES);
  deg_k<<<gEdge, 256, 0, stream>>>(e_dst, dis);
  dis_k<<<gNode, 256, 0, stream>>>(dis);
  coef_k<<<gEdge, 256, 0, stream>>>(e_src, e_dst, dis, coef);

  // --- 4 GCN layers ---
  const float* layer_in = x;
  int K = FIN;
  for (int l = 0; l < 4; ++l) {
    gemm_wmma_f32<<<gGemm, 256, 0, stream>>>(layer_in, W[l], bufA, NNODES, K);
    agg_init_k<<<gElem, 256, 0, stream>>>(bufA, dis, b[l], bufB);
    agg_edges_k<<<gEdge16, 256, 0, stream>>>(bufA, e_src, e_dst, coef, bufB);
    zero_k<<<1, 128, 0, stream>>>(stats, 128);
    bn_stats_k<<<256, 256, 0, stream>>>(bufB, stats, NNODES);
    bn_coeffs_k<<<1, 64, 0, stream>>>(stats, g[l], bt[l], coeffs);
    bn_apply_k<<<gElem, 256, 0, stream>>>(bufB, coeffs);
    layer_in = bufB;
    K = HID;
  }

  // --- pool + fc + log_softmax ---
  zero_k<<<(NGRAPH * HID + NGRAPH + 255) / 256, 256, 0, stream>>>(psum, NGRAPH * HID + NGRAPH);
  pool_k<<<gNode16, 256, 0, stream>>>(bufB, batch, psum, pcnt);
  fc_logsoftmax_k<<<(NGRAPH + 255) / 256, 256, 0, stream>>>(psum, pcnt, fcW, fcb, out);
}